// MinRegressionCombinationLoss_41034117546050
// MI455X (gfx1250) — compile-verified
//
#include <hip/hip_runtime.h>
#include <hip/hip_bf16.h>

// ---------------------------------------------------------------------------
// MinRegressionCombinationLoss on MI455X (gfx1250)
//
// output, multilabels: [B=32768, C=1000] f32.  Per row:
//   o = sigmoid(x); lp = log(o+eps); ln = log(1-o+eps); gain = lp - ln
//   base = -sum(ln); pos_gain = sum(gain where label==1 && gain>0)
//   best  = max(gain where label==1)
//   per_sample = base - (pos_gain > 0 ? pos_gain : best)   // any_pos <=> pos_gain>0
// result = mean(per_sample)
//
// Memory-bound: 262 MB streamed -> ~11 us @ 23.3 TB/s. TDM streams tiles to
// LDS (double-buffered, TENSORcnt-synced); v_wmma_f32_16x16x4_f32 with an
// all-ones B matrix performs the two row-sum reductions in f32 for free.
// Inner loop runs entirely in log2 units on raw v_exp_f32/v_log_f32/v_rcp_f32
// (args are never denormal/inf: eps=1e-12 >> FLT_MIN, inputs finite), and the
// single ln2 scale is applied once per row in the finalize.
// ---------------------------------------------------------------------------

typedef __attribute__((ext_vector_type(2))) float        v2f;
typedef __attribute__((ext_vector_type(8))) float        v8f;
typedef __attribute__((ext_vector_type(4))) unsigned int u32x4;
typedef __attribute__((ext_vector_type(8))) int          i32x8;
typedef __attribute__((ext_vector_type(4))) int          i32x4;

#ifndef __has_builtin
#define __has_builtin(x) 0
#endif
#if __has_builtin(__builtin_amdgcn_tensor_load_to_lds) && \
    __has_builtin(__builtin_amdgcn_s_wait_tensorcnt)
#define HAVE_TDM 1
#else
#define HAVE_TDM 0
#endif

#define C_DIM        1000
#define ROWS_PER_WG  64            // 4 waves x 16 rows
#define COLS         40            // 1000 = 25 * 40, divisible by 4 (WMMA K)
#define NCHUNK       (C_DIM / COLS)
#define THREADS      128

#define LN2F    0.69314718055994531f
#define LOG2EF  1.44269504088896341f

// Raw hardware transcendentals (1 instruction each); hedge for toolchains
// where the builtins are absent.
#if __has_builtin(__builtin_amdgcn_exp2f)
__device__ __forceinline__ float hw_exp2(float x) { return __builtin_amdgcn_exp2f(x); }
#else
__device__ __forceinline__ float hw_exp2(float x) { return exp2f(x); }
#endif
#if __has_builtin(__builtin_amdgcn_logf)
__device__ __forceinline__ float hw_log2(float x) { return __builtin_amdgcn_logf(x); }
#else
__device__ __forceinline__ float hw_log2(float x) { return log2f(x); }
#endif
#if __has_builtin(__builtin_amdgcn_rcpf)
__device__ __forceinline__ float hw_rcp(float x) { return __builtin_amdgcn_rcpf(x); }
#else
__device__ __forceinline__ float hw_rcp(float x) { return 1.0f / x; }
#endif

// All outputs in log2 units; caller scales by ln2 once per row.
__device__ __forceinline__ void elem_math(float x, float t,
                                          float& o_ln, float& o_pos, float& o_mx) {
  const float EPSF = 1e-12f;
  float e = hw_exp2(x * -LOG2EF);             // exp(-x): v_mul + v_exp_f32
  float s = 1.0f + e;
  float r = hw_rcp(s);                        // v_rcp_f32
  float o = r * fmaf(-s, r, 2.0f);            // Newton step -> ~1 ulp sigmoid
  // Reference's naive forms preserved (incl. (1-o) cancellation -> log(eps)
  // clamp when o rounds to 1.0f).
  float lp2 = hw_log2(o + EPSF);              // log2(o+eps)
  float ln2v = hw_log2((1.0f - o) + EPSF);    // log2(1-o+eps)
  float g  = lp2 - ln2v;                      // gain / ln2  (sign preserved)
  bool isT = (t == 1.0f);
  o_ln  = ln2v;
  o_pos = (isT && (g > 0.0f)) ? g : 0.0f;
  o_mx  = isT ? g : -__builtin_inff();
}

#if HAVE_TDM
// 2D TDM descriptor per CDNA5 ISA ch.8: group0 = {count|flags, lds_addr,
// global_addr, type=2}; group1 packs data_size=4B, tensor_dim0/1, tile_dim0/1,
// tensor_dim0_stride. Groups 2/3 zero (2D tensor). This toolchain's builtin
// takes 6 args (g0:v4u, g1:v8i, g2:v4i, g3:v4i, g4:v8i, cpol).
__device__ __forceinline__ void tdm_load_tile(const float* gsrc, unsigned lds_off,
                                              int nrows_total) {
  unsigned long long ga = (unsigned long long)(const void*)gsrc;
  u32x4 g0;
  g0[0] = 1u;                                              // count=1, user mode
  g0[1] = lds_off;                                         // LDS byte address
  g0[2] = (unsigned)ga;                                    // global_addr[31:0]
  g0[3] = ((unsigned)(ga >> 32) & 0x01FFFFFFu) | (2u << 30); // addr[56:32] | type=2
  i32x8 g1;
  g1[0] = (int)(2u << 16);                                  // data_size = 4 bytes
  g1[1] = (int)(((unsigned)C_DIM & 0xFFFFu) << 16);         // tensor_dim0 lo16
  g1[2] = (int)((((unsigned)C_DIM >> 16) & 0xFFFFu) |
                (((unsigned)nrows_total & 0xFFFFu) << 16)); // dim0 hi | dim1 lo
  g1[3] = (int)((((unsigned)nrows_total >> 16) & 0xFFFFu) |
                ((unsigned)COLS << 16));                    // dim1 hi | tile_dim0
  g1[4] = (int)(unsigned)ROWS_PER_WG;                       // tile_dim1 (tile_dim2=0)
  g1[5] = (int)(unsigned)C_DIM;                             // tensor_dim0_stride lo32
  g1[6] = 0;                                                // stride hi | dim1_stride
  g1[7] = 0;
  i32x4 gz4 = {0, 0, 0, 0};
  i32x8 gz8 = {0, 0, 0, 0, 0, 0, 0, 0};
  __builtin_amdgcn_tensor_load_to_lds(g0, g1, gz4, gz4, gz8, 0);
}
#endif

__global__ __launch_bounds__(THREADS)
void row_loss_kernel(const float* __restrict__ gout, const float* __restrict__ glab,
                     float* __restrict__ ps, int B) {
  __shared__ float tile[2][2][ROWS_PER_WG * COLS];   // [buffer][array]  40 KB
  __shared__ float redB[ROWS_PER_WG];
  __shared__ float redP[ROWS_PER_WG];

  const int tid  = threadIdx.x;
  const int lane = tid & 31;
  const int wave = tid >> 5;
  const int m    = lane & 15;        // WMMA A-matrix row (M)
  const int kh   = lane >> 4;        // K half: lanes 0-15 -> K0,1; 16-31 -> K2,3
  const int rowInTile = wave * 16 + m;
  const int rowBase   = blockIdx.x * ROWS_PER_WG;

  v8f accB = {0.f, 0.f, 0.f, 0.f, 0.f, 0.f, 0.f, 0.f};   // sum log2(1-o+eps)
  v8f accP = {0.f, 0.f, 0.f, 0.f, 0.f, 0.f, 0.f, 0.f};   // sum positive gains /ln2
  float mx = -__builtin_inff();                           // max gain /ln2 (true lbls)
  const v2f bones = {1.0f, 1.0f};                         // B = all-ones -> row sums

#if HAVE_TDM
  if (wave == 0) {   // one wave drives the TDM pipeline for the workgroup
    const size_t off0 = (size_t)rowBase * C_DIM;
    tdm_load_tile(gout + off0, (unsigned)(size_t)&tile[0][0][0], B);
    tdm_load_tile(glab + off0, (unsigned)(size_t)&tile[0][1][0], B);
  }
#endif

  for (int c = 0; c < NCHUNK; ++c) {
    const int buf = c & 1;
#if HAVE_TDM
    if (wave == 0) {
      __builtin_amdgcn_s_wait_tensorcnt(0);   // chunk c landed in LDS
    }
    __syncthreads();                          // publish to all waves; also
                                              // guarantees buf^1 readers of
                                              // chunk c-1 are done
    if (wave == 0 && (c + 1) < NCHUNK) {      // prefetch chunk c+1
      const size_t off = (size_t)rowBase * C_DIM + (size_t)(c + 1) * COLS;
      tdm_load_tile(gout + off, (unsigned)(size_t)&tile[buf ^ 1][0][0], B);
      tdm_load_tile(glab + off, (unsigned)(size_t)&tile[buf ^ 1][1][0], B);
    }
#else
    for (int idx = tid; idx < ROWS_PER_WG * COLS; idx += THREADS) {
      int r = idx / COLS, col = idx - r * COLS;
      size_t g = (size_t)(rowBase + r) * C_DIM + (size_t)c * COLS + col;
      tile[buf][0][idx] = gout[g];
      tile[buf][1][idx] = glab[g];
    }
    __syncthreads();
#endif

    const int lb = rowInTile * COLS;
#pragma unroll
    for (int k0 = 0; k0 < COLS; k0 += 4) {
      const int cb = lb + k0 + 2 * kh;
      float x0 = tile[buf][0][cb], x1 = tile[buf][0][cb + 1];
      float t0 = tile[buf][1][cb], t1 = tile[buf][1][cb + 1];
      float l0, p0, gm0, l1, p1, gm1;
      elem_math(x0, t0, l0, p0, gm0);
      elem_math(x1, t1, l1, p1, gm1);
      v2f aB = {l0, l1};
      v2f aP = {p0, p1};
      // D = A x ones + C  ->  per-row f32 sums accumulated in C operand
      accB = __builtin_amdgcn_wmma_f32_16x16x4_f32(
          false, aB, false, bones, (short)0, accB, false, false);
      accP = __builtin_amdgcn_wmma_f32_16x16x4_f32(
          false, aP, false, bones, (short)0, accP, false, false);
      mx = fmaxf(mx, fmaxf(gm0, gm1));
    }
  }

  // Row max: combine the two K-halves of each row (lanes m and m+16).
  mx = fmaxf(mx, __shfl_xor(mx, 16, 32));

  // C/D layout: lane 0 VGPR i -> row i; lane 16 VGPR i -> row 8+i (any column).
  if (lane == 0) {
#pragma unroll
    for (int i = 0; i < 8; ++i) { redB[wave * 16 + i] = accB[i]; redP[wave * 16 + i] = accP[i]; }
  } else if (lane == 16) {
#pragma unroll
    for (int i = 0; i < 8; ++i) { redB[wave * 16 + 8 + i] = accB[i]; redP[wave * 16 + 8 + i] = accP[i]; }
  }
  __syncthreads();

  if (lane < 16) {
    float sB = redB[wave * 16 + lane];        // log2 units
    float sP = redP[wave * 16 + lane];        // log2 units
    float base = -LN2F * sB;
    // any_pos <=> sum of strictly-positive gains > 0 (scale-invariant test)
    float max_gain = (sP > 0.0f) ? (LN2F * sP) : (LN2F * mx);
    ps[rowBase + wave * 16 + lane] = base - max_gain;
  }
}

// Deterministic mean over B per-sample losses (fixed-order tree, no atomics).
__global__ __launch_bounds__(1024)
void mean_kernel(const float* __restrict__ ps, float* __restrict__ out, int n) {
  __shared__ float red[1024];
  float s = 0.0f;
  for (int i = threadIdx.x; i < n; i += 1024) s += ps[i];
  red[threadIdx.x] = s;
  __syncthreads();
  for (int off = 512; off > 0; off >>= 1) {
    if ((int)threadIdx.x < off) red[threadIdx.x] += red[threadIdx.x + off];
    __syncthreads();
  }
  if (threadIdx.x == 0) out[0] = red[0] / (float)n;
}

extern "C" void kernel_launch(void* const* d_in, const int* in_sizes, int n_in,
                              void* d_out, int out_size, void* d_ws, size_t ws_size,
                              hipStream_t stream) {
  const float* gout = (const float*)d_in[0];   // "output"      [B, 1000] f32
  const float* glab = (const float*)d_in[1];   // "multilabels" [B, 1000] f32
  float* res = (float*)d_out;                  // scalar f32
  float* ps  = (float*)d_ws;                   // per-sample losses [B]

  const int B = in_sizes[0] / C_DIM;           // 32768
  const int blocks = B / ROWS_PER_WG;          // 512

  row_loss_kernel<<<blocks, THREADS, 0, stream>>>(gout, glab, ps, B);
  mean_kernel<<<1, 1024, 0, stream>>>(ps, res, B);
}